// HyperbolicDistanceProbe_90993177133165
// MI455X (gfx1250) — compile-verified
//
#include <hip/hip_runtime.h>
#include <math.h>

// ---------------------------------------------------------------------------
// HyperbolicDistanceProbe on gfx1250 (MI455X), fp32 WMMA (V_WMMA_F32_16X16X4_F32)
// B=8, S=2048, D=1024, R=128.
//
// All B-side GEMM operands are stored "K-permuted": within each group of 4
// K-values the storage order is [k, k+2, k+1, k+3].  With that layout the
// B-fragment for lane-half kh is the contiguous pair at offset (k0 + 2*kh) --
// identical addressing to the A-fragment -- so every WMMA operand load is a
// single global_load_b64.
// ---------------------------------------------------------------------------

typedef float v2f __attribute__((ext_vector_type(2)));
typedef float v8f __attribute__((ext_vector_type(8)));

#define HD_EPS      1e-15f
#define HD_SQEPS    3.1622776601683794e-8f   /* sqrt(1e-15) */
#define HD_ATCLIP   (1.0f - 1e-7f)
#define HD_MAXSQ    (1.0f - 1e-5f)
#define HD_SQMAXSQ  0.9999949999874999f      /* sqrt(1-1e-5) */

#define NB   8
#define NS   2048
#define ND   1024
#define NR   128
#define NROWS (NB * NS)   /* 16384 */

__device__ __forceinline__ v8f wmma_f32x4(v2f a, v2f b, v8f c) {
    // D = A(16x4,f32) * B(4x16,f32) + C(16x16,f32)
    return __builtin_amdgcn_wmma_f32_16x16x4_f32(
        /*neg_a=*/false, a, /*neg_b=*/false, b,
        /*c_mod=*/(short)0, c, /*reuse_a=*/false, /*reuse_b=*/false);
}

// permuted column index: groups of 4, order [k, k+2, k+1, k+3]
__device__ __forceinline__ int permc(int c) {
    int j = c & 3;
    return (c & ~3) | (((j & 1) << 1) | (j >> 1));
}

// ---------------------------------------------------------------------------
// Prep: copy W[rows][K] -> Wp with K-permuted columns.
// ---------------------------------------------------------------------------
__global__ __launch_bounds__(256)
void hd_permK(const float* __restrict__ W, float* __restrict__ Wp,
              int total, int kmask) {
    int i = blockIdx.x * 256 + threadIdx.x;
    if (i >= total) return;
    int c = i & kmask;
    Wp[(i & ~kmask) | permc(c)] = W[i];
}

// ---------------------------------------------------------------------------
// Kernel 1: fused  x = E*P^T -> expmap0 -> project -> Mx = p*M^T -> mobius
//           scaling -> project -> res, resp (K-permuted), sq.
// One block = 16 rows x 128 cols; 8 wave32, wave w owns cols [16w, 16w+16).
// ---------------------------------------------------------------------------
__global__ __launch_bounds__(256)
void hd_stage1(const float* __restrict__ E,     // [NROWS, ND]
               const float* __restrict__ Pp,    // [NR, ND]  K-permuted
               const float* __restrict__ Mp,    // [NR, NR]  K-permuted
               float* __restrict__ res,         // [NROWS, NR] row-major
               float* __restrict__ resp,        // [NROWS, NR] K-permuted
               float* __restrict__ sqout) {     // [NROWS]
    __shared__ __align__(16) float xs[16][132];
    __shared__ float rowsum[16];
    __shared__ float rowscale[16];
    __shared__ float rownorm[16];

    const int tid  = threadIdx.x;
    const int lane = tid & 31;
    const int wave = tid >> 5;
    const int l16  = lane & 15;
    const int kh   = lane >> 4;          // 0 for lanes 0-15, 1 for lanes 16-31
    const int m0   = blockIdx.x * 16;
    const int n0   = wave * 16;

    if (tid < 16) rowsum[tid] = 0.0f;

    // ---------------- GEMM1: x[m, n] = sum_d E[m,d] * P[n,d] ----------------
    v8f acc = {};
    const float* arow = E  + (size_t)(m0 + l16) * ND;
    const float* brow = Pp + (size_t)(n0 + l16) * ND;
    #pragma unroll 8
    for (int k0 = 0; k0 < ND; k0 += 4) {
        if ((k0 & 127) == 0) __builtin_prefetch(arow + k0 + 256, 0, 1);
        v2f a = *(const v2f*)(arow + k0 + 2 * kh);   // A[l16, k0+2kh .. +1]
        v2f b = *(const v2f*)(brow + k0 + 2 * kh);   // permuted: {k0+kh, k0+2+kh}
        acc = wmma_f32x4(a, b, acc);
    }

    // scatter acc tile into LDS: vgpr i -> row i + 8*kh, col n0 + l16
    #pragma unroll
    for (int i = 0; i < 8; ++i)
        xs[i + 8 * kh][n0 + l16] = acc[i];
    __syncthreads();

    // ---- row sums of x^2: 256-thread partials + LDS float atomics ----------
    {
        int r = tid >> 4, c = tid & 15;
        float s = 0.0f;
        #pragma unroll
        for (int i = 0; i < 8; ++i) { float v = xs[r][c + 16 * i]; s += v * v; }
        atomicAdd(&rowsum[r], s);                       // ds_add_f32
    }
    __syncthreads();

    // ------------- expmap0 + project (per-row scalars) ----------------------
    if (tid < 16) {
        float s   = rowsum[tid];
        rowsum[tid] = 0.0f;                             // re-arm for round 2
        float xn  = sqrtf(fmaxf(s, HD_EPS));
        float th  = tanhf(xn);
        float pn  = fmaxf(th, HD_SQEPS);                // ||p|| clamped
        float f   = fminf(1.0f, HD_SQMAXSQ / pn);       // ball projection
        rowscale[tid] = (th / xn) * f;                  // x -> p scale
        float pfin = th * f;
        rownorm[tid]  = fmaxf(pfin, HD_SQEPS);          // pn used by mobius
    }
    __syncthreads();

    // apply scale in LDS: xs now holds p
    for (int idx = tid; idx < 16 * NR; idx += 256) {
        int r = idx >> 7, c = idx & (NR - 1);
        xs[r][c] *= rowscale[r];
    }
    __syncthreads();

    // ---------------- GEMM2: Mx[m, q] = sum_r p[m,r] * M[q,r] ---------------
    v8f acc2 = {};
    const float* brow2 = Mp + (size_t)(n0 + l16) * NR;
    #pragma unroll 8
    for (int k0 = 0; k0 < NR; k0 += 4) {
        v2f a = *(const v2f*)(&xs[l16][k0 + 2 * kh]);   // ds_load_b64
        v2f b = *(const v2f*)(brow2 + k0 + 2 * kh);     // permuted b64
        acc2 = wmma_f32x4(a, b, acc2);
    }
    __syncthreads();   // done reading p from LDS; safe to overwrite

    #pragma unroll
    for (int i = 0; i < 8; ++i)
        xs[i + 8 * kh][n0 + l16] = acc2[i];
    __syncthreads();

    // ---- row sums of Mx^2 ---------------------------------------------------
    {
        int r = tid >> 4, c = tid & 15;
        float s = 0.0f;
        #pragma unroll
        for (int i = 0; i < 8; ++i) { float v = xs[r][c + 16 * i]; s += v * v; }
        atomicAdd(&rowsum[r], s);
    }
    __syncthreads();

    // --------- mobius_matvec scaling + projection (per-row) -----------------
    if (tid < 16) {
        float mxn = sqrtf(fmaxf(rowsum[tid], HD_EPS));
        float pn  = rownorm[tid];
        float t   = tanhf(mxn / pn * atanhf(fminf(pn, HD_ATCLIP)));
        float rn  = fmaxf(t, HD_SQEPS);
        float f2  = fminf(1.0f, HD_SQMAXSQ / rn);
        rowscale[tid] = (t / mxn) * f2;                 // Mx -> res scale
        float rfin = t * f2;
        sqout[m0 + tid] = fminf(rfin * rfin, HD_MAXSQ); // clipped ||res||^2
    }
    __syncthreads();

    // write res (row-major) and resp (K-permuted), both coalesced-ish
    for (int idx = tid; idx < 16 * NR; idx += 256) {
        int r = idx >> 7, c = idx & (NR - 1);
        float v = xs[r][c] * rowscale[r];
        size_t rowbase = (size_t)(m0 + r) * NR;
        res [rowbase + c]        = v;
        resp[rowbase + permc(c)] = v;
    }
}

// ---------------------------------------------------------------------------
// Kernel 2: all-pairs Poincare distance per batch.
// Block = 64x64 output tile, 8 waves; wave w: col subtile (w&3), two row
// subtiles at 16*(w>>2) and 16*(w>>2)+32.  K = 128 -> 32 fp32-WMMA steps.
// A-frags from row-major res, B-frags from K-permuted resp: all b64 loads.
// ---------------------------------------------------------------------------
__global__ __launch_bounds__(256)
void hd_dist(const float* __restrict__ res,    // [NB*NS, NR] row-major
             const float* __restrict__ resp,   // [NB*NS, NR] K-permuted
             const float* __restrict__ sqv,    // [NB*NS]
             float* __restrict__ dist) {       // [NB, NS, NS]
    const int b  = blockIdx.z;
    const int i0 = blockIdx.y * 64;
    const int j0 = blockIdx.x * 64;
    const float* rb  = res  + (size_t)b * NS * NR;
    const float* rbp = resp + (size_t)b * NS * NR;
    const float* sb  = sqv  + (size_t)b * NS;
    float*       db  = dist + (size_t)b * NS * NS;

    const int lane = threadIdx.x & 31;
    const int wave = threadIdx.x >> 5;
    const int l16  = lane & 15;
    const int kh   = lane >> 4;

    const int jt  = j0 + (wave & 3) * 16;           // column tile base
    const int it0 = i0 + (wave >> 2) * 16;          // row tile 0 base
    const int it1 = it0 + 32;                       // row tile 1 base

    const float* browp = rbp + (size_t)(jt  + l16) * NR;
    const float* arow0 = rb  + (size_t)(it0 + l16) * NR;
    const float* arow1 = rb  + (size_t)(it1 + l16) * NR;

    v8f acc0 = {};
    v8f acc1 = {};
    #pragma unroll 8
    for (int k0 = 0; k0 < NR; k0 += 4) {
        v2f bfrag = *(const v2f*)(browp + k0 + 2 * kh);  // permuted b64
        v2f a0    = *(const v2f*)(arow0 + k0 + 2 * kh);
        v2f a1    = *(const v2f*)(arow1 + k0 + 2 * kh);
        acc0 = wmma_f32x4(a0, bfrag, acc0);
        acc1 = wmma_f32x4(a1, bfrag, acc1);
    }

    // epilogue: dist = 2*asinh( sqrt( d2 / ((1-sq_i)(1-sq_j)) ) )
    const int   col = jt + l16;
    const float sqj = sb[col];
    const float omj = 1.0f - sqj;

    #pragma unroll
    for (int i = 0; i < 8; ++i) {
        {
            int   row = it0 + i + 8 * kh;
            float sqi = sb[row];
            float d2  = fmaxf(sqi + sqj - 2.0f * acc0[i], 0.0f);
            float den = fmaxf((1.0f - sqi) * omj, HD_EPS);
            float out = (d2 > 0.0f) ? 2.0f * asinhf(sqrtf(d2 / den)) : 0.0f;
            db[(size_t)row * NS + col] = out;
        }
        {
            int   row = it1 + i + 8 * kh;
            float sqi = sb[row];
            float d2  = fmaxf(sqi + sqj - 2.0f * acc1[i], 0.0f);
            float den = fmaxf((1.0f - sqi) * omj, HD_EPS);
            float out = (d2 > 0.0f) ? 2.0f * asinhf(sqrtf(d2 / den)) : 0.0f;
            db[(size_t)row * NS + col] = out;
        }
    }
}

// ---------------------------------------------------------------------------
extern "C" void kernel_launch(void* const* d_in, const int* in_sizes, int n_in,
                              void* d_out, int out_size, void* d_ws, size_t ws_size,
                              hipStream_t stream) {
    (void)in_sizes; (void)n_in; (void)out_size; (void)ws_size;
    const float* E = (const float*)d_in[0];   // embeddings [8,2048,1024]
    const float* P = (const float*)d_in[1];   // proj       [128,1024]
    const float* M = (const float*)d_in[2];   // mobius     [128,128]
    float* out = (float*)d_out;               // dist       [8,2048,2048]

    float* res   = (float*)d_ws;                        // 16384*128
    float* resp  = res   + (size_t)NROWS * NR;          // 16384*128
    float* projp = resp  + (size_t)NROWS * NR;          // 128*1024
    float* mobp  = projp + (size_t)NR * ND;             // 128*128
    float* sqv   = mobp  + (size_t)NR * NR;             // 16384

    hd_permK<<<(NR * ND + 255) / 256, 256, 0, stream>>>(P, projp, NR * ND, ND - 1);
    hd_permK<<<(NR * NR + 255) / 256, 256, 0, stream>>>(M, mobp,  NR * NR, NR - 1);

    hd_stage1<<<NROWS / 16, 256, 0, stream>>>(E, projp, mobp, res, resp, sqv);

    dim3 grid(NS / 64, NS / 64, NB);
    hd_dist<<<grid, 256, 0, stream>>>(res, resp, sqv, out);
}